// SilhouetteScore_32676111188224
// MI455X (gfx1250) — compile-verified
//
#include <hip/hip_runtime.h>

typedef float v2f __attribute__((ext_vector_type(2)));
typedef float v8f __attribute__((ext_vector_type(8)));

#define N_PTS        8192
#define DIMS         128
#define NUM_CLU      64
#define KSTEPS       (DIMS / 4)   // 32 wmma_16x16x4 steps cover K=128
#define TM           32           // rows per block
#define TNJ          64           // cols per iteration (4 col-subtiles x 16)

// ws layout (floats): [0,8192) sq ; [8192,8256) counts ; [8256] accum
#define WS_SQ        0
#define WS_CNT       N_PTS
#define WS_ACC       (N_PTS + NUM_CLU)

__global__ void sil_init_ws(float* ws) {
  int t = threadIdx.x;
  if (t < NUM_CLU) ws[WS_CNT + t] = 0.0f;
  if (t == 0)      ws[WS_ACC]     = 0.0f;
}

__global__ void sil_norms_counts(const float* __restrict__ x,
                                 const int* __restrict__ labels,
                                 float* ws) {
  int i = blockIdx.x * blockDim.x + threadIdx.x;
  if (i >= N_PTS) return;
  const float4* row = (const float4*)(x + (size_t)i * DIMS);
  float s = 0.0f;
#pragma unroll
  for (int k = 0; k < DIMS / 4; ++k) {
    float4 v = row[k];
    s += v.x * v.x + v.y * v.y + v.z * v.z + v.w * v.w;
  }
  ws[WS_SQ + i] = s;
  atomicAdd(&ws[WS_CNT + labels[i]], 1.0f);
}

__global__ __launch_bounds__(256) void sil_main(const float* __restrict__ x,
                                                const int* __restrict__ labels,
                                                float* ws) {
  __shared__ float S[TM][NUM_CLU];   // per-block segment sums (8 KB)
  for (int idx = threadIdx.x; idx < TM * NUM_CLU; idx += 256)
    (&S[0][0])[idx] = 0.0f;
  __syncthreads();

  const float* sq = ws + WS_SQ;
  const int lane    = threadIdx.x & 31;
  const int wave    = threadIdx.x >> 5;
  const int waveRow = (wave & 1) * 16;   // 0 or 16  -> 32 rows
  const int waveCol = (wave >> 1) * 16;  // 0..48    -> 64 cols per iter
  const int row0    = blockIdx.x * TM;

  // WMMA f32 16x16x4 A-fragment addressing (ISA 7.12.2, 32-bit A 16x4):
  //   lane L holds A[row=L%16][k0], A[row][k0+1] with k0 = (L/16)*2
  const int mrow = lane & 15;
  const int koff = (lane >> 4) * 2;
  const int mhi  = (lane >> 4) << 3;     // +8 row offset for lanes 16..31 in C/D

  // Load this wave's A fragments once (16 rows x full K=128) -> 64 VGPRs
  v2f afrag[KSTEPS];
  const float* arow = x + (size_t)(row0 + waveRow + mrow) * DIMS + koff;
#pragma unroll
  for (int s = 0; s < KSTEPS; ++s)
    afrag[s] = *(const v2f*)(arow + s * 4);

  float sqi[8];
#pragma unroll
  for (int e = 0; e < 8; ++e)
    sqi[e] = sq[row0 + waveRow + e + mhi];

  for (int col0 = 0; col0 < N_PTS; col0 += TNJ) {
    const int j = col0 + waveCol + mrow;           // this lane's column
    const float* brow = x + (size_t)j * DIMS + koff;  // B = X^T: same pattern as A
    v8f c = {};
#pragma unroll
    for (int s = 0; s < KSTEPS; ++s) {
      v2f b = *(const v2f*)(brow + s * 4);
      // D = A x B + C  (f32 matrix pipe, full precision)
      c = __builtin_amdgcn_wmma_f32_16x16x4_f32(
          /*neg_a=*/false, afrag[s], /*neg_b=*/false, b,
          /*c_mod=*/(short)0, c, /*reuse_a=*/false, /*reuse_b=*/false);
    }
    const float sqj = sq[j];
    const int   lj  = labels[j];
#pragma unroll
    for (int e = 0; e < 8; ++e) {
      float d2   = sqi[e] + sqj - 2.0f * c[e];
      float dist = (d2 > 0.0f) ? sqrtf(d2) : 0.0f;   // grad-safe sqrt semantics
      atomicAdd(&S[waveRow + e + mhi][lj], dist);     // ds_add_f32 segment-sum
    }
  }
  __syncthreads();

  // Epilogue: one thread per row computes a, b, score; accumulate globally.
  if (threadIdx.x < TM) {
    const int r  = threadIdx.x;
    const int i  = row0 + r;
    const int li = labels[i];
    const float* counts = ws + WS_CNT;
    float own = counts[li];
    float a   = S[r][li] / fmaxf(own - 1.0f, 1.0f);
    float b   = 3.402823466e38f;  // finfo(f32).max
    for (int cix = 0; cix < NUM_CLU; ++cix) {
      float cnt = counts[cix];
      if (cix == li || cnt == 0.0f) continue;   // mask own + empty clusters
      b = fminf(b, S[r][cix] / cnt);
    }
    float score = (own > 1.0f) ? (b - a) / fmaxf(b, a) : 0.0f;
    atomicAdd(ws + WS_ACC, score);
  }
}

__global__ void sil_finalize(const float* ws, float* out) {
  out[0] = ws[WS_ACC] / (float)N_PTS;
}

extern "C" void kernel_launch(void* const* d_in, const int* in_sizes, int n_in,
                              void* d_out, int out_size, void* d_ws, size_t ws_size,
                              hipStream_t stream) {
  (void)in_sizes; (void)n_in; (void)out_size; (void)ws_size;
  const float* x      = (const float*)d_in[0];
  const int*   labels = (const int*)d_in[1];
  float*       ws     = (float*)d_ws;
  float*       out    = (float*)d_out;

  sil_init_ws<<<1, 128, 0, stream>>>(ws);
  sil_norms_counts<<<N_PTS / 256, 256, 0, stream>>>(x, labels, ws);
  sil_main<<<N_PTS / TM, 256, 0, stream>>>(x, labels, ws);
  sil_finalize<<<1, 1, 0, stream>>>(ws, out);
}